// UnbatchedMinkowski_13537736917142
// MI455X (gfx1250) — compile-verified
//
#include <hip/hip_runtime.h>
#include <hip/hip_bf16.h>

// ---- static problem geometry (matches reference) ----
#define N_TOTAL 500000
#define NCH     128        // C
#define NGRP    32         // G
#define CG      4          // C/G
#define NB_CLD  8          // B (number of clouds)
#define DEPTH   4          // async gather pipeline depth (ring slots per wave)

// cumulative boundaries of LENGTHS = [40000,55000,70000,45000,80000,65000,90000,55000]
__device__ __forceinline__ int cloud_of(int i) {
    int c = 0;
    c += (i >= 40000);
    c += (i >= 95000);
    c += (i >= 165000);
    c += (i >= 210000);
    c += (i >= 290000);
    c += (i >= 355000);
    c += (i >= 445000);
    return c;
}

// CDNA5 async global->LDS gather: each lane moves 16B, one wave = 512B row.
// Tracked by ASYNCcnt; completes in order within the wave.
__device__ __forceinline__ void async_gather_b128(unsigned lds_byte_addr, const float4* gaddr) {
    asm volatile("global_load_async_to_lds_b128 %0, %1, off"
                 :: "v"(lds_byte_addr), "v"(gaddr)
                 : "memory");
}

template <int N>
__device__ __forceinline__ void wait_asynccnt() {
    asm volatile("s_wait_asynccnt %0" :: "i"(N) : "memory");
}

// ---------------------------------------------------------------------------
// Pass 1: per-block partial sums/sumsq per (cloud, group).
// Wave-per-row; lane l covers channels 4l..4l+3 == group l (wave32 sweet spot).
// Rows are gathered through a 4-deep async LDS ring (ASYNCcnt pipeline).
// Block partial layout: partial[block][cloud][group][2] = [nb][8][32][2] f32.
// ---------------------------------------------------------------------------
__global__ __launch_bounds__(256) void gn_partial_kernel(
    const float4* __restrict__ feats4,   // [N_TOTAL*32] float4 (32 float4 per row)
    const int*    __restrict__ perm,     // [N_TOTAL]
    float*        __restrict__ partial)  // [gridDim.x * 512]
{
    // 16KB: first used as per-wave async ring (8 waves * 4 slots * 128 floats),
    // then reused for the deterministic cross-wave reduction [wave][cloud][group][2].
    __shared__ float smem[4096];

    const int lane = threadIdx.x & 31;
    const int wave = threadIdx.x >> 5;
    const int W    = gridDim.x * 8;                // total waves
    const int gw   = blockIdx.x * 8 + wave;        // gw < 8192 <= N_TOTAL always

    const unsigned ringBase =
        (unsigned)(size_t)(&smem[wave * 512]) + (unsigned)(lane * 16);

    float s[NB_CLD], ss[NB_CLD];
#pragma unroll
    for (int b = 0; b < NB_CLD; ++b) { s[b] = 0.f; ss[b] = 0.f; }

    const int i0    = gw;
    const int niter = (N_TOTAL - 1 - i0) / W + 1;  // >= DEPTH by grid construction

    // prologue: fill the ring with DEPTH in-flight gathers
#pragma unroll
    for (int j = 0; j < DEPTH; ++j) {
        const int r = perm[i0 + j * W];
        async_gather_b128(ringBase + (unsigned)(j * 512),
                          feats4 + (size_t)r * 32 + lane);
    }

    auto consume = [&](int k) {
        const float4 v =
            *(const float4*)&smem[wave * 512 + (k & 3) * 128 + lane * 4]; // ds_load_b128
        const int cl = cloud_of(i0 + k * W);
        const float rs  = (v.x + v.y) + (v.z + v.w);
        const float rss = (v.x * v.x + v.y * v.y) + (v.z * v.z + v.w * v.w);
#pragma unroll
        for (int b = 0; b < NB_CLD; ++b) {         // branchless; keeps s/ss in VGPRs
            const bool m = (cl == b);
            s[b]  += m ? rs  : 0.f;
            ss[b] += m ? rss : 0.f;
        }
    };

    // steady state: 4 gathers always in flight; oldest guaranteed done at <=3
    int k = 0;
    for (; k < niter - DEPTH; ++k) {
        wait_asynccnt<DEPTH - 1>();
        consume(k);
        const int rn = perm[i0 + (k + DEPTH) * W];
        async_gather_b128(ringBase + (unsigned)((k & 3) * 512),
                          feats4 + (size_t)rn * 32 + lane);
    }
    // drain
    wait_asynccnt<3>(); consume(k + 0);
    wait_asynccnt<2>(); consume(k + 1);
    wait_asynccnt<1>(); consume(k + 2);
    wait_asynccnt<0>(); consume(k + 3);

    // write per-wave partials into (our own) 512-float slice, then cross-wave reduce
#pragma unroll
    for (int b = 0; b < NB_CLD; ++b) {
        smem[((wave * NB_CLD + b) * NGRP + lane) * 2 + 0] = s[b];
        smem[((wave * NB_CLD + b) * NGRP + lane) * 2 + 1] = ss[b];
    }
    __syncthreads();

    // 256 threads -> t = (cloud c, group g); deterministic fixed-order sum over 8 waves
    const int t = threadIdx.x;
    const int c = t >> 5;
    const int g = t & 31;
    float as = 0.f, ass = 0.f;
#pragma unroll
    for (int w = 0; w < 8; ++w) {
        as  += smem[((w * NB_CLD + c) * NGRP + g) * 2 + 0];
        ass += smem[((w * NB_CLD + c) * NGRP + g) * 2 + 1];
    }
    partial[(size_t)blockIdx.x * 512 + t * 2 + 0] = as;
    partial[(size_t)blockIdx.x * 512 + t * 2 + 1] = ass;
}

// ---------------------------------------------------------------------------
// Pass 2: single block, 256 threads = (cloud, group). Deterministic serial
// reduction over block partials, then mean / rsqrt(var + eps).
// stats layout: stats[(cloud*32 + group)*2 + {0:mean, 1:inv}]
// ---------------------------------------------------------------------------
__global__ __launch_bounds__(256) void gn_stats_kernel(
    const float* __restrict__ partial,
    float*       __restrict__ stats,
    int nb)
{
    const int t = threadIdx.x;            // 0..255
    float s = 0.f, ss = 0.f;
    for (int b = 0; b < nb; ++b) {
        s  += partial[(size_t)b * 512 + t * 2 + 0];
        ss += partial[(size_t)b * 512 + t * 2 + 1];
    }
    // cnt = LENGTHS[cloud] * Cg
    const float cnts[NB_CLD] = {160000.f, 220000.f, 280000.f, 180000.f,
                                320000.f, 260000.f, 360000.f, 220000.f};
    const float cnt  = cnts[t >> 5];
    const float mean = s / cnt;
    const float var  = ss / cnt - mean * mean;
    const float inv  = rsqrtf(var + 1e-5f);
    stats[t * 2 + 0] = mean;
    stats[t * 2 + 1] = inv;
}

// ---------------------------------------------------------------------------
// Pass 3: normalize + affine. Same async-ring gather; lane l = group l.
// Per-lane preload of (mean, inv) for all 8 clouds -> per-row branchless select.
// ---------------------------------------------------------------------------
__global__ __launch_bounds__(256) void gn_norm_kernel(
    const float4* __restrict__ feats4,
    const int*    __restrict__ perm,
    const float4* __restrict__ gamma4,   // [32] float4
    const float4* __restrict__ beta4,    // [32] float4
    const float*  __restrict__ stats,    // [256*2]
    float4*       __restrict__ out4)     // [N_TOTAL*32]
{
    __shared__ float smem[4096];         // 8 waves * 4 slots * 128 floats = 16KB ring

    const int lane = threadIdx.x & 31;
    const int wave = threadIdx.x >> 5;
    const int W    = gridDim.x * 8;
    const int gw   = blockIdx.x * 8 + wave;

    const unsigned ringBase =
        (unsigned)(size_t)(&smem[wave * 512]) + (unsigned)(lane * 16);

    const float4 ga = gamma4[lane];
    const float4 be = beta4[lane];

    float mean[NB_CLD], inv[NB_CLD];
#pragma unroll
    for (int b = 0; b < NB_CLD; ++b) {
        mean[b] = stats[(b * NGRP + lane) * 2 + 0];
        inv[b]  = stats[(b * NGRP + lane) * 2 + 1];
    }

    const int i0    = gw;
    const int niter = (N_TOTAL - 1 - i0) / W + 1;  // >= DEPTH by grid construction

    // row ids for in-flight slots, oldest-first shift registers
    int r0 = perm[i0 + 0 * W];
    async_gather_b128(ringBase + 0 * 512, feats4 + (size_t)r0 * 32 + lane);
    int r1 = perm[i0 + 1 * W];
    async_gather_b128(ringBase + 1 * 512, feats4 + (size_t)r1 * 32 + lane);
    int r2 = perm[i0 + 2 * W];
    async_gather_b128(ringBase + 2 * 512, feats4 + (size_t)r2 * 32 + lane);
    int r3 = perm[i0 + 3 * W];
    async_gather_b128(ringBase + 3 * 512, feats4 + (size_t)r3 * 32 + lane);

    auto emit = [&](int k, int row) {
        const float4 v =
            *(const float4*)&smem[wave * 512 + (k & 3) * 128 + lane * 4]; // ds_load_b128
        const int cl = cloud_of(i0 + k * W);
        float m = 0.f, iv = 0.f;
#pragma unroll
        for (int b = 0; b < NB_CLD; ++b) {
            const bool q = (cl == b);
            m  += q ? mean[b] : 0.f;
            iv += q ? inv[b]  : 0.f;
        }
        float4 y;
        y.x = (v.x - m) * iv * ga.x + be.x;
        y.y = (v.y - m) * iv * ga.y + be.y;
        y.z = (v.z - m) * iv * ga.z + be.z;
        y.w = (v.w - m) * iv * ga.w + be.w;
        out4[(size_t)row * 32 + lane] = y;          // global_store_b128
    };

    int k = 0;
    for (; k < niter - DEPTH; ++k) {
        wait_asynccnt<DEPTH - 1>();
        emit(k, r0);
        r0 = r1; r1 = r2; r2 = r3;
        r3 = perm[i0 + (k + DEPTH) * W];
        async_gather_b128(ringBase + (unsigned)((k & 3) * 512),   // (k+4)&3 == k&3
                          feats4 + (size_t)r3 * 32 + lane);
    }
    wait_asynccnt<3>(); emit(k + 0, r0);
    wait_asynccnt<2>(); emit(k + 1, r1);
    wait_asynccnt<1>(); emit(k + 2, r2);
    wait_asynccnt<0>(); emit(k + 3, r3);
}

// ---------------------------------------------------------------------------
extern "C" void kernel_launch(void* const* d_in, const int* in_sizes, int n_in,
                              void* d_out, int out_size, void* d_ws, size_t ws_size,
                              hipStream_t stream) {
    const float* feats = (const float*)d_in[0];   // [N, C] f32
    const float* gamma = (const float*)d_in[1];   // [C]    f32
    const float* beta  = (const float*)d_in[2];   // [C]    f32
    const int*   perm  = (const int*)d_in[3];     // [N]    int
    float*       out   = (float*)d_out;           // [N, C] f32
    float*       ws    = (float*)d_ws;

    (void)in_sizes; (void)n_in; (void)out_size;

    // workspace budget: nb1 block partials of 512 floats + 512 floats of stats
    int nb1 = 768;
    {
        const size_t avail_f = ws_size / sizeof(float);
        if (avail_f < (size_t)(nb1 + 1) * 512) {
            long cap = (long)(avail_f / 512) - 1;
            nb1 = (cap < 1) ? 1 : (int)cap;
        }
    }
    float* partial = ws;
    float* stats   = ws + (size_t)nb1 * 512;

    const float4* feats4 = (const float4*)feats;
    const float4* gamma4 = (const float4*)gamma;
    const float4* beta4  = (const float4*)beta;
    float4*       out4   = (float4*)out;

    gn_partial_kernel<<<nb1, 256, 0, stream>>>(feats4, perm, partial);
    gn_stats_kernel<<<1, 256, 0, stream>>>(partial, stats, nb1);
    gn_norm_kernel<<<1024, 256, 0, stream>>>(feats4, perm, gamma4, beta4, stats, out4);
}